// OpticalDepth_91027536872144
// MI455X (gfx1250) — compile-verified
//
#include <hip/hip_runtime.h>
#include <hip/hip_bf16.h>
#include <stdint.h>

// ---------------------------------------------------------------------------
// OpticalDepth: out[8][N][30]
//   tau[n][j][k]   = relu(c_sel[n][j] * w_gas[j][k])        j = 0..7
//   ke[n][g]       = 4-layer MLP(t_p[n]) per gas g = 0..5   (2->6->4->4->1)
//   out[j]         = tau[j]                                  j = 0,1
//   out[j]         = tau[j] * FILTERS[j-2] * ke[j-2]         j = 2..7
// Store-bandwidth-bound: 960 B out / 44 B in per point -> ~45 us floor @23.3TB/s.
// CDNA5 paths: async global->LDS weight staging (ASYNCcnt), LDS broadcasts,
// NT b64 stores, and layer-1 of the MLP on V_WMMA_F32_16X16X4_F32 with a
// run-time probe that discovers the (undocumented) f32 B-operand lane layout.
// ---------------------------------------------------------------------------

#define NCH 30
#define WG  256

typedef __attribute__((ext_vector_type(2))) float v2f;
typedef __attribute__((ext_vector_type(8))) float v8f;
typedef __attribute__((address_space(1))) int* gint_p;
typedef __attribute__((address_space(3))) int* lint_p;

#if __has_builtin(__builtin_amdgcn_wmma_f32_16x16x4_f32)
#define HAVE_WMMA_F32X4 1
#endif

// FILTERS rows as 30-bit masks (bit k == FILTERS[row][k]); compile-time folded.
__device__ constexpr unsigned kFilt[6] = {
    0x303FFFFFu, 0x3F3F0003u, 0x30003333u, 0x3C3FC003u, 0x30000013u, 0x30000CCFu
};

// LDS layout (float offsets)
#define O_WGAS 0      // 8*30  = 240
#define O_W1   240    // 6*2*6 = 72
#define O_B1   312    // 6*6   = 36
#define O_W2   348    // 6*6*4 = 144
#define O_B2   492    // 6*4   = 24
#define O_W3   516    // 6*4*4 = 96
#define O_B3   612    // 6*4   = 24
#define O_W4   636    // 6*4   = 24
#define O_B4   660    // 6
#define NWTOT  666

__device__ __forceinline__ void stage_async(float* ldst, const float* gsrc) {
#if __has_builtin(__builtin_amdgcn_global_load_async_to_lds_b32)
  __builtin_amdgcn_global_load_async_to_lds_b32(
      (gint_p)(uintptr_t)gsrc, (lint_p)ldst, /*offset=*/0, /*cpol=*/0);
#else
  *ldst = *gsrc;
#endif
}

__global__ __launch_bounds__(WG) void OpticalDepth_91027536872144_kernel(
    const float* __restrict__ t_p, const float* __restrict__ c,
    const float* __restrict__ w_gas,
    const float* __restrict__ W1, const float* __restrict__ b1,
    const float* __restrict__ W2, const float* __restrict__ b2,
    const float* __restrict__ W3, const float* __restrict__ b3,
    const float* __restrict__ W4, const float* __restrict__ b4,
    float* __restrict__ out, int npts) {
  __shared__ float wl[NWTOT];
#ifdef HAVE_WMMA_F32X4
  __shared__ int inv[64];   // B-slot code -> (k<<4)|n
#endif

  // ---- Stage all weights into LDS: one predicated async load per array ----
  {
    const int t = threadIdx.x;
    if (t < 240) stage_async(&wl[O_WGAS + t], w_gas + t);
    if (t < 72)  stage_async(&wl[O_W1 + t], W1 + t);
    if (t < 36)  stage_async(&wl[O_B1 + t], b1 + t);
    if (t < 144) stage_async(&wl[O_W2 + t], W2 + t);
    if (t < 24)  stage_async(&wl[O_B2 + t], b2 + t);
    if (t < 96)  stage_async(&wl[O_W3 + t], W3 + t);
    if (t < 24)  stage_async(&wl[O_B3 + t], b3 + t);
    if (t < 24)  stage_async(&wl[O_W4 + t], W4 + t);
    if (t < 6)   stage_async(&wl[O_B4 + t], b4 + t);
  }
#if __has_builtin(__builtin_amdgcn_s_wait_asynccnt)
  __builtin_amdgcn_s_wait_asynccnt(0);
#else
  asm volatile("s_wait_asynccnt 0" ::: "memory");
#endif
  __syncthreads();

  const int lane = threadIdx.x & 31;
  const int n    = blockIdx.x * WG + threadIdx.x;
  const int nl   = (n < npts) ? n : (npts - 1);  // clamp loads; stores guarded

  // ---- Inputs (no early return: EXEC must stay all-ones for WMMA) ----
  const float x0 = t_p[2 * nl + 0];
  const float x1 = t_p[2 * nl + 1];

  float cs[8];
  {
    const float* cr = c + (size_t)nl * 9;
    cs[0] = cr[0]; cs[1] = cr[1]; cs[2] = cr[2]; cs[3] = cr[3];
    cs[4] = cr[4]; cs[5] = cr[6]; cs[6] = cr[7]; cs[7] = cr[8];  // skip col 5
  }

#ifdef HAVE_WMMA_F32X4
  // ---- Probe: discover the f32 WMMA B-operand layout (exact for codes<64) --
  // A = 16x4 identity in the documented A layout; B slots carry their code.
  {
    v2f aI, bP;
    aI.x = (lane == 0 || lane == 18) ? 1.0f : 0.0f;  // K=0 (lo) / K=2 (hi)
    aI.y = (lane == 1 || lane == 19) ? 1.0f : 0.0f;  // K=1 (lo) / K=3 (hi)
    bP.x = (float)lane;
    bP.y = (float)(lane + 32);
    v8f z = {};
    v8f dP = __builtin_amdgcn_wmma_f32_16x16x4_f32(false, aI, false, bP,
                                                   (short)0, z, false, false);
    if (lane < 16) {  // D[m][n] = B[m][n] for m<4; documented D layout
      inv[(int)dP[0]] = (0 << 4) | lane;
      inv[(int)dP[1]] = (1 << 4) | lane;
      inv[(int)dP[2]] = (2 << 4) | lane;
      inv[(int)dP[3]] = (3 << 4) | lane;
    }
  }
  __syncthreads();
  const int e0 = inv[lane], e1 = inv[lane + 32];
  const int k0 = e0 >> 4, n0 = e0 & 15;
  const int k1 = e1 >> 4, n1 = e1 & 15;

  // ---- B = t_p activations for the wave's two 16-point tiles (gas-invariant)
  v2f bT[2];
#pragma unroll
  for (int t = 0; t < 2; ++t) {
    float s00 = __shfl(x0, n0 + 16 * t, 32);
    float s01 = __shfl(x1, n0 + 16 * t, 32);
    float s10 = __shfl(x0, n1 + 16 * t, 32);
    float s11 = __shfl(x1, n1 + 16 * t, 32);
    bT[t].x = (k0 == 0) ? s00 : (k0 == 1) ? s01 : 0.0f;  // zero unused slots
    bT[t].y = (k1 == 0) ? s10 : (k1 == 1) ? s11 : 0.0f;
  }
#endif

  // ---- 6 tiny MLPs; layer 1 on the matrix pipe, layers 2-4 scalar ----
  float ke[6];
#pragma unroll
  for (int g = 0; g < 6; ++g) {
    float h1[6];
#ifdef HAVE_WMMA_F32X4
    {
      const int m = lane & 15;
      // A = W1^T (M=out feat, K=in feat), documented A layout; K=2,3 zero.
      v2f aW;
      aW.x = (lane < 16) ? wl[O_W1 + g * 12 + 0 + m] : 0.0f;
      aW.y = (lane < 16) ? wl[O_W1 + g * 12 + 6 + m] : 0.0f;
      v8f cb;  // bias broadcast along points (rows m>=6 are dead)
#pragma unroll
      for (int r = 0; r < 6; ++r) cb[r] = wl[O_B1 + g * 6 + r];
      cb[6] = 0.0f; cb[7] = 0.0f;
      v8f d0 = __builtin_amdgcn_wmma_f32_16x16x4_f32(false, aW, false, bT[0],
                                                     (short)0, cb, false, false);
      v8f d1 = __builtin_amdgcn_wmma_f32_16x16x4_f32(false, aW, false, bT[1],
                                                     (short)0, cb, false, false);
#pragma unroll
      for (int j = 0; j < 6; ++j) {
        float sw = __shfl_xor(d1[j], 16, 32);      // tile1: point n at lane n
        float v  = (lane < 16) ? d0[j] : sw;       // own point's feature j
        h1[j] = fmaxf(v, 0.0f);
      }
    }
#else
#pragma unroll
    for (int j = 0; j < 6; ++j) {
      float a = wl[O_B1 + g * 6 + j];
      a = fmaf(x0, wl[O_W1 + g * 12 + 0 + j], a);
      a = fmaf(x1, wl[O_W1 + g * 12 + 6 + j], a);
      h1[j] = fmaxf(a, 0.0f);
    }
#endif
    float h2[4];
#pragma unroll
    for (int o = 0; o < 4; ++o) {
      float a = wl[O_B2 + g * 4 + o];
#pragma unroll
      for (int f = 0; f < 6; ++f)
        a = fmaf(h1[f], wl[O_W2 + g * 24 + f * 4 + o], a);
      h2[o] = fmaxf(a, 0.0f);
    }
    float h3[4];
#pragma unroll
    for (int o = 0; o < 4; ++o) {
      float a = wl[O_B3 + g * 4 + o];
#pragma unroll
      for (int f = 0; f < 4; ++f)
        a = fmaf(h2[f], wl[O_W3 + g * 16 + f * 4 + o], a);
      h3[o] = fmaxf(a, 0.0f);
    }
    float a = wl[O_B4 + g];
#pragma unroll
    for (int f = 0; f < 4; ++f)
      a = fmaf(h3[f], wl[O_W4 + g * 4 + f], a);
    ke[g] = fmaxf(a, 0.0f);
  }

  // ---- tau, filter, scale; non-temporal b64 stores (rows are 8B aligned) ----
  if (n < npts) {
    const size_t plane = (size_t)npts * NCH;
    float* rowbase = out + (size_t)n * NCH;
#pragma unroll
    for (int j = 0; j < 8; ++j) {
      const unsigned mask = (j < 2) ? 0x3FFFFFFFu : kFilt[j - 2];
      const float scale = (j < 2) ? 1.0f : ke[j - 2];
      float buf[NCH];
#pragma unroll
      for (int k = 0; k < NCH; ++k) {
        if ((mask >> k) & 1u) {
          float t = fmaxf(cs[j] * wl[O_WGAS + j * NCH + k], 0.0f);
          buf[k] = t * scale;
        } else {
          buf[k] = 0.0f;
        }
      }
      float* rb = rowbase + (size_t)j * plane;
#pragma unroll
      for (int k = 0; k < NCH / 2; ++k) {
        v2f p = {buf[2 * k], buf[2 * k + 1]};
        __builtin_nontemporal_store(p, (v2f*)(rb + 2 * k));
      }
    }
  }
}

extern "C" void kernel_launch(void* const* d_in, const int* in_sizes, int n_in,
                              void* d_out, int out_size, void* d_ws, size_t ws_size,
                              hipStream_t stream) {
  (void)n_in; (void)out_size; (void)d_ws; (void)ws_size;
  const float* t_p   = (const float*)d_in[0];
  const float* c     = (const float*)d_in[1];
  const float* w_gas = (const float*)d_in[2];
  const float* W1    = (const float*)d_in[3];
  const float* b1    = (const float*)d_in[4];
  const float* W2    = (const float*)d_in[5];
  const float* b2    = (const float*)d_in[6];
  const float* W3    = (const float*)d_in[7];
  const float* b3    = (const float*)d_in[8];
  const float* W4    = (const float*)d_in[9];
  const float* b4    = (const float*)d_in[10];
  float* out = (float*)d_out;

  const int npts = in_sizes[0] / 2;  // t_p is (N, 2)
  dim3 grid((npts + WG - 1) / WG), block(WG);
  hipLaunchKernelGGL(OpticalDepth_91027536872144_kernel, grid, block, 0, stream,
                     t_p, c, w_gas, W1, b1, W2, b2, W3, b3, W4, b4, out, npts);
}